// ConvCapsLayer3D_22814866276894
// MI455X (gfx1250) — compile-verified
//
#include <hip/hip_runtime.h>
#include <cmath>

typedef __attribute__((ext_vector_type(2))) float v2f;
typedef __attribute__((ext_vector_type(8))) float v8f;
typedef __attribute__((ext_vector_type(4))) unsigned int v4u;
typedef __attribute__((ext_vector_type(8))) int v8i;
typedef __attribute__((ext_vector_type(4))) int v4i;

#define EPSF 1e-8f

// Problem constants (from reference): b=8, ch_i=32, n_i=8, H=W=32, ch_j=32, n_j=4, oc=128
// Vote tensor layout: X[b][pix(1024)][ci(32)][oc(128)]  where oc = cj*4 + nj
// Logits layout:      B[b][ci][pix(1024)][cj(32)]       (softmax set = contiguous 32768)
// S_hat layout:       Sh[b][pix][cj*4+nj]

// Workspace layout in floats
#define X_OFF    0ul
#define X_SZ     (8ul * 1024 * 32 * 128)   // 33,554,432 floats (134 MB)
#define B_OFF    (X_OFF + X_SZ)
#define B_SZ     (8ul * 32 * 1024 * 32)    //  8,388,608 floats (33.5 MB)
#define SH_OFF   (B_OFF + B_SZ)
#define SH_SZ    (8ul * 1024 * 128)        //  1,048,576 floats
#define RMAX_OFF (SH_OFF + SH_SZ)
#define RINV_OFF (RMAX_OFF + 256)
// total ~ 42,992,128 floats = ~164 MB of d_ws

// ---------------------------------------------------------------------------
// TDM: 1-D contiguous copy of `nelem` dwords from global to LDS.
// D# per CDNA5 ISA §8.3/8.4: group0 = {count=1 | lds_addr | global_addr | type=2},
// group1 = {data_size=4B, tensor_dim0 = tile_dim0 = nelem, tensor_dim1 = 1,
//           tile_dim1/2 = 0 (1-D)}, groups 2/3 = 0. Tracked by TENSORcnt.
// This toolchain exposes the 6-arg builtin (g0, g1, g2, g3, g4, cpol).
// ---------------------------------------------------------------------------
__device__ __forceinline__ void tdm_load_1d(unsigned lds_byte_addr,
                                            const void* gptr, unsigned nelem) {
  const unsigned long long ga = (unsigned long long)(uintptr_t)gptr;
  v4u g0;
  g0[0] = 1u;                                           // count=1, user mode
  g0[1] = lds_byte_addr;                                // bits 63:32  lds_addr
  g0[2] = (unsigned)(ga & 0xFFFFFFFFu);                 // bits 95:64  global_addr lo
  g0[3] = (unsigned)((ga >> 32) & 0x01FFFFFFu)          // bits 120:96 global_addr hi
          | (2u << 30);                                 // bits 127:126 type=2 (image)
  v8i g1;
  g1[0] = (int)(2u << 16);                              // data_size = 2 -> 4 bytes
  g1[1] = (int)((nelem & 0xFFFFu) << 16);               // tensor_dim0[15:0] @ bits 63:48
  g1[2] = (int)(((nelem >> 16) & 0xFFFFu) | (1u << 16));// tensor_dim0[31:16] | tensor_dim1=1
  g1[3] = (int)((nelem & 0xFFFFu) << 16);               // tile_dim0 @ bits 127:112
  g1[4] = 0;                                            // tile_dim1=0, tile_dim2=0 (1-D)
  g1[5] = (int)nelem;                                   // tensor_dim0_stride lo
  g1[6] = 0;
  g1[7] = 0;
  const v4i z4 = {0, 0, 0, 0};
  const v8i z8 = {0, 0, 0, 0, 0, 0, 0, 0};
  __builtin_amdgcn_tensor_load_to_lds(g0, g1, z4, z4, z8, 0);
}

// ---------------------------------------------------------------------------
// init: zero routing logits and the entropy output slot
// ---------------------------------------------------------------------------
__global__ __launch_bounds__(256) void init_kernel(float* __restrict__ Bws,
                                                   float* __restrict__ ent_slot) {
  size_t i = (size_t)blockIdx.x * blockDim.x + threadIdx.x;
  size_t stride = (size_t)gridDim.x * blockDim.x;
  for (size_t j = i; j < B_SZ; j += stride) Bws[j] = 0.0f;
  if (i == 0) *ent_slot = 0.0f;
}

// ---------------------------------------------------------------------------
// Conv3d (stride-8 depth => 256 independent 128x1024x27 GEMMs) via f32 WMMA.
// Grid: (16 pixel-tiles of 64px, ci=32, b=8). Block: 256 threads = 8 waves.
// Wave w owns oc rows [16w,16w+16); 4 N-tiles of 16 pixels.
// Weights arrive via TDM (contiguous [oc][27]); K handled as 6 full WMMA
// steps + 1 tail step with the k=27 pad zeroed in the A fragment.
// ---------------------------------------------------------------------------
__global__ __launch_bounds__(256) void conv_wmma_kernel(
    const float* __restrict__ in, const float* __restrict__ cw,
    const float* __restrict__ cb, float* __restrict__ X) {
  __shared__ __align__(16) float sWf[128 * 27];  // weights [oc][k], contiguous (TDM)
  __shared__ float sP[28 * 72];                  // im2col [k][px], stride 72 (bank-safe)
  __shared__ float sO[64 * 129];                 // output staging [px][oc]

  const int pt = blockIdx.x;  // 64-pixel tile (2 rows of y)
  const int ci = blockIdx.y;
  const int b  = blockIdx.z;
  const int t  = threadIdx.x;

  // Wave 0 kicks off the async weight DMA; everyone builds im2col meanwhile.
  if (t < 32) {
    tdm_load_1d((unsigned)(uintptr_t)(void*)sWf, cw, 128 * 27);
  }

  // Build im2col patch: only n_i slices 0..2 of this ci are ever touched.
  const int y0 = pt * 2;
  const float* inb = in + ((size_t)b * 32 + ci) * 8 * 1024;  // [ni][y][x]
  for (int i = t; i < 27 * 64; i += 256) {
    int k = i / 64, px = i % 64;
    int dz = k / 9, r = k % 9;
    int dy = r / 3, dx = r % 3;
    int ly = px >> 5, x = px & 31;
    int yy = y0 + ly + dy - 1;
    int xx = x + dx - 1;
    float v = 0.0f;
    if (yy >= 0 && yy < 32 && xx >= 0 && xx < 32) v = inb[dz * 1024 + yy * 32 + xx];
    sP[k * 72 + px] = v;
  }
  if (t < 64) sP[27 * 72 + t] = 0.0f;  // K pad row for the tail WMMA step

  if (t < 32) __builtin_amdgcn_s_wait_tensorcnt(0);
  __syncthreads();

  const int w  = t >> 5;   // wave id -> M tile
  const int l  = t & 31;   // lane
  const int lh = l & 15;
  const int hi = l >> 4;   // 0: K pair {k0,k0+1}; 1: K pair {k0+2,k0+3}

  const float* wrow = &sWf[(w * 16 + lh) * 27];
  v8f acc[4] = {};
#pragma unroll
  for (int k0 = 0; k0 < 24; k0 += 4) {
    v2f av;
    av[0] = wrow[k0 + hi * 2 + 0];
    av[1] = wrow[k0 + hi * 2 + 1];
#pragma unroll
    for (int nt = 0; nt < 4; ++nt) {
      const int bcol = nt * 16 + lh;
      v2f bv;
      bv[0] = sP[(k0 + hi * 2 + 0) * 72 + bcol];
      bv[1] = sP[(k0 + hi * 2 + 1) * 72 + bcol];
      acc[nt] = __builtin_amdgcn_wmma_f32_16x16x4_f32(
          false, av, false, bv, (short)0, acc[nt], false, false);
    }
  }
  {  // tail: K = 24,25,26,27(pad). hi==1 lane pair {26,27}: zero the pad lane.
    v2f av;
    av[0] = wrow[24 + hi * 2];
    av[1] = hi ? 0.0f : wrow[25];
#pragma unroll
    for (int nt = 0; nt < 4; ++nt) {
      const int bcol = nt * 16 + lh;
      v2f bv;
      bv[0] = sP[(24 + hi * 2 + 0) * 72 + bcol];
      bv[1] = sP[(24 + hi * 2 + 1) * 72 + bcol];  // row 27 is zeroed
      acc[nt] = __builtin_amdgcn_wmma_f32_16x16x4_f32(
          false, av, false, bv, (short)0, acc[nt], false, false);
    }
  }

  // Stage D fragments (+bias) to LDS:  D vgpr r, lane l -> M=r+8*hi, N=lane%16
#pragma unroll
  for (int nt = 0; nt < 4; ++nt) {
    const int px = nt * 16 + lh;
#pragma unroll
    for (int r = 0; r < 8; ++r) {
      const int oc = w * 16 + hi * 8 + r;
      sO[px * 129 + oc] = acc[nt][r] + cb[oc];
    }
  }
  __syncthreads();

  // Coalesced store: X[b][pix][ci][oc], 128 contiguous floats per pixel.
  float* Xb = X + (((size_t)b * 1024 + (size_t)y0 * 32) * 32 + ci) * 128;
  for (int i = t; i < 64 * 128; i += 256) {
    const int pxl = i >> 7, oc = i & 127;
    Xb[(size_t)pxl * 4096 + oc] = sO[pxl * 129 + oc];
  }
}

// ---------------------------------------------------------------------------
// Per-(b,ci) softmax statistics over the contiguous 32768-element logit row.
// ---------------------------------------------------------------------------
__global__ __launch_bounds__(256) void rowstats_kernel(const float* __restrict__ Bws,
                                                       float* __restrict__ rmax,
                                                       float* __restrict__ rinv) {
  const int row = blockIdx.x;  // b*32+ci
  const float* p = Bws + (size_t)row * 32768;
  const int t = threadIdx.x;
  __shared__ float red[256];

  float m = -INFINITY;
  for (int i = t; i < 32768; i += 256) m = fmaxf(m, p[i]);
  red[t] = m;
  __syncthreads();
  for (int s = 128; s > 0; s >>= 1) {
    if (t < s) red[t] = fmaxf(red[t], red[t + s]);
    __syncthreads();
  }
  const float rm = red[0];
  __syncthreads();

  float sum = 0.0f;
  for (int i = t; i < 32768; i += 256) sum += __expf(p[i] - rm);
  red[t] = sum;
  __syncthreads();
  for (int s = 128; s > 0; s >>= 1) {
    if (t < s) red[t] += red[t + s];
    __syncthreads();
  }
  if (t == 0) {
    rmax[row] = rm;
    rinv[row] = 1.0f / red[0];
  }
}

// ---------------------------------------------------------------------------
// One routing round per (b, pixel): S = sum_ci k*x, squash, optional B update.
// Block = 128 threads; thread t owns (cj = t>>2, nj = t&3).
// The 16KB contiguous vote block is TDM'd into LDS (overlapped with the
// softmax k-tile construction) and consumed twice from LDS.
// ---------------------------------------------------------------------------
__global__ __launch_bounds__(128) void routing_kernel(
    const float* __restrict__ X, float* Bws, const float* __restrict__ rmax,
    const float* __restrict__ rinv, float* __restrict__ Sh, int do_update) {
  const int bp = blockIdx.x;  // b*1024 + pix
  const int b = bp >> 10, pix = bp & 1023;
  const int t = threadIdx.x;

  __shared__ __align__(16) float sX[4096];  // votes [ci][cj*4+nj] (TDM)
  __shared__ float sk[32 * 32];             // coupling k[ci][cj]
  __shared__ float ssh[128];                // S_hat[cj*4+nj]

  // Wave 0 starts the async vote DMA; all waves compute the coupling tile.
  if (t < 32) {
    tdm_load_1d((unsigned)(uintptr_t)(void*)sX, X + (size_t)bp * 4096, 4096);
  }

  const float* Bb = Bws + ((size_t)b * 32) * 32768 + (size_t)pix * 32;
  for (int i = t; i < 1024; i += 128) {
    const int ci = i >> 5;
    const int row = b * 32 + ci;
    sk[i] = __expf(Bb[(size_t)ci * 32768 + (i & 31)] - rmax[row]) * rinv[row];
  }

  if (t < 32) __builtin_amdgcn_s_wait_tensorcnt(0);
  __syncthreads();

  const int cj = t >> 2, nj = t & 3;
  float s = 0.0f;
#pragma unroll 8
  for (int ci = 0; ci < 32; ++ci) s += sk[ci * 32 + cj] * sX[ci * 128 + cj * 4 + nj];

  // squash over a size-1 axis:  |s| * s / (1 + s^2 + eps)
  const float sh = fabsf(s) * s / (1.0f + s * s + EPSF);
  Sh[(size_t)bp * 128 + t] = sh;

  if (do_update) {
    ssh[t] = sh;
    __syncthreads();
    for (int i = t; i < 1024; i += 128) {
      const int ci = i >> 5, cjj = i & 31;
      const float4 xv = *(const float4*)(&sX[ci * 128 + cjj * 4]);
      const float acc = ssh[cjj * 4 + 0] * xv.x + ssh[cjj * 4 + 1] * xv.y +
                        ssh[cjj * 4 + 2] * xv.z + ssh[cjj * 4 + 3] * xv.w;
      Bws[((size_t)(b * 32 + ci)) * 32768 + (size_t)pix * 32 + cjj] += acc;
    }
  }
}

// ---------------------------------------------------------------------------
// out[b][cj][nj][pix] = Sh[b][pix][cj*4+nj]
// ---------------------------------------------------------------------------
__global__ __launch_bounds__(256) void out_kernel(const float* __restrict__ Sh,
                                                  float* __restrict__ out) {
  const int i = blockIdx.x * 256 + threadIdx.x;  // 0 .. 262143
  if (i >= 8 * 128 * 1024) return;
  const int pix = i & 1023;
  const int r = i >> 10;  // b*128 + cj*4 + nj
  const int b = r >> 7;
  const int cjnj = r & 127;
  out[i] = Sh[((size_t)(b * 1024 + pix)) * 128 + cjnj];
}

// ---------------------------------------------------------------------------
// Entropy of final coupling coefficients, normalized by uniform entropy.
// ---------------------------------------------------------------------------
__global__ __launch_bounds__(256) void entropy_kernel(
    const float* __restrict__ Bws, const float* __restrict__ rmax,
    const float* __restrict__ rinv, float* __restrict__ ent_slot) {
  const int row = blockIdx.x;  // b*32+ci
  const float* p = Bws + (size_t)row * 32768;
  const float rm = rmax[row], ri = rinv[row];
  const int t = threadIdx.x;
  float acc = 0.0f;
  for (int i = t; i < 32768; i += 256) {
    const float pk = __expf(p[i] - rm) * ri;
    acc -= pk * __logf(pk + EPSF);
  }
  __shared__ float red[256];
  red[t] = acc;
  __syncthreads();
  for (int s = 128; s > 0; s >>= 1) {
    if (t < s) red[t] += red[t + s];
    __syncthreads();
  }
  if (t == 0) {
    // mean over 256 rows of ent / ln(32768)
    const float scale = 1.0f / (256.0f * 10.397207708399179f);
    atomicAdd(ent_slot, red[0] * scale);
  }
}

// ---------------------------------------------------------------------------
extern "C" void kernel_launch(void* const* d_in, const int* in_sizes, int n_in,
                              void* d_out, int out_size, void* d_ws, size_t ws_size,
                              hipStream_t stream) {
  const float* inputs = (const float*)d_in[0];  // (8,32,8,32,32)
  const float* conv_w = (const float*)d_in[1];  // (128,1,3,3,3)
  const float* conv_b = (const float*)d_in[2];  // (128,)
  float* out = (float*)d_out;                   // 1048576 (out) + 1 (entropy)

  float* ws   = (float*)d_ws;
  float* X    = ws + X_OFF;
  float* Bws  = ws + B_OFF;
  float* Sh   = ws + SH_OFF;
  float* rmax = ws + RMAX_OFF;
  float* rinv = ws + RINV_OFF;
  float* ent  = out + 8 * 128 * 1024;

  init_kernel<<<2048, 256, 0, stream>>>(Bws, ent);

  conv_wmma_kernel<<<dim3(16, 32, 8), 256, 0, stream>>>(inputs, conv_w, conv_b, X);

  for (int it = 0; it < 3; ++it) {
    rowstats_kernel<<<256, 256, 0, stream>>>(Bws, rmax, rinv);
    routing_kernel<<<8192, 128, 0, stream>>>(X, Bws, rmax, rinv, Sh,
                                             /*do_update=*/(it < 2) ? 1 : 0);
  }

  out_kernel<<<1024, 256, 0, stream>>>(Sh, out);
  entropy_kernel<<<256, 256, 0, stream>>>(Bws, rmax, rinv, ent);
}